// VectorQuantizer_28759101014242
// MI455X (gfx1250) — compile-verified
//
#include <hip/hip_runtime.h>
#include <hip/hip_bf16.h>

// VQ-VAE eval forward for MI455X (gfx1250), fp32 WMMA 16x16x4 path.
//
// Outputs (concatenated in d_out, float):
//   [0 .. 2097151]        z_q  (16,8,128,128)
//   [2097152]             commit_loss (scalar)
//   [2097153 .. 2359296]  indices (16,128,128) stored as float values

typedef float v2f __attribute__((ext_vector_type(2)));
typedef float v8f __attribute__((ext_vector_type(8)));

#define NUM_EMB   1024
#define EMB_DIM   8
#define HW        16384          // 128*128
#define N_ROWS    262144         // 16*128*128
#define N_TILES   16384          // N_ROWS / 16
#define BLOCKS    1024
#define TPB       256
#define WPB       8              // waves per block (wave32)
#define TOT_WAVES (BLOCKS * WPB) // 8192
#define ITERS     (N_TILES / TOT_WAVES) // 2
#define ZQ_ELEMS  2097152
#define LOSS_OFF  2097152
#define IDX_OFF   2097153

__global__ __launch_bounds__(TPB) void vq_main_kernel(
    const float* __restrict__ z, const float* __restrict__ wgt,
    float* __restrict__ out, float* __restrict__ blockLoss) {
  // --- LDS: codebook pre-shuffled into WMMA B-matrix layout -----------------
  // B (4x16 f32, per k-tile): lane = code column; lanes 0-15 hold K(d)=base+0/1
  // in VGPR0/1, lanes 16-31 hold K(d)=base+2/3.  Stored as float2 per lane.
  __shared__ float2 sBlo[64 * 32];   // d0..d3   16 KB
  __shared__ float2 sBhi[64 * 32];   // d4..d7   16 KB
  __shared__ float  sWn[NUM_EMB];    // ||w||^2   4 KB
  __shared__ int    sIdx[WPB * 16];
  __shared__ float  sMin[WPB * 16];
  __shared__ float  sLoss[WPB];

  const int tid  = threadIdx.x;
  const int lane = tid & 31;
  const int wib  = tid >> 5;                  // wave in block
  const int waveId = blockIdx.x * WPB + wib;  // global wave id

  // --- Stage codebook: 1024 codes / 256 threads = 4 each --------------------
#pragma unroll
  for (int j = 0; j < 4; ++j) {
    int k = tid + j * TPB;
    const float4* wp = (const float4*)(wgt + (size_t)k * EMB_DIM);
    float4 lo = wp[0];
    float4 hi = wp[1];
    int base = ((k >> 4) << 5) + (k & 15);
    sBlo[base]      = make_float2(lo.x, lo.y);
    sBlo[base + 16] = make_float2(lo.z, lo.w);
    sBhi[base]      = make_float2(hi.x, hi.y);
    sBhi[base + 16] = make_float2(hi.z, hi.w);
    sWn[k] = lo.x*lo.x + lo.y*lo.y + lo.z*lo.z + lo.w*lo.w
           + hi.x*hi.x + hi.y*hi.y + hi.z*hi.z + hi.w*hi.w;
  }
  __syncthreads();

  const int m    = lane & 15;        // row within 16-row tile / column id
  const int db   = (lane >> 4) << 1; // channel base for A layout (0 or 2)
  float waveLoss = 0.0f;

  for (int it = 0; it < ITERS; ++it) {
    const int t  = waveId + it * TOT_WAVES;   // row-tile index
    const int n0 = t << 4;
    const int b   = n0 >> 14;
    const int rem = n0 & 16383;
    const int h   = rem >> 7;
    const int w0  = rem & 127;                // multiple of 16

    // --- A tiles: 16 rows x (d0..3) and (d4..7), fp32 WMMA A layout ---------
    const float* zp = z + (((size_t)(b * 8 + db) * 128 + h) * 128 + (w0 + m));
    v2f alo, ahi;
    alo.x = zp[0];
    alo.y = zp[HW];
    ahi.x = zp[4 * HW];
    ahi.y = zp[5 * HW];
    float np = alo.x*alo.x + alo.y*alo.y + ahi.x*ahi.x + ahi.y*ahi.y;
    float rowNorm = np + __shfl_xor(np, 16, 32);  // lane L holds ||z_row L||^2

    float minv[8];
    int   mini[8];
#pragma unroll
    for (int r = 0; r < 8; ++r) { minv[r] = 3.4e38f; mini[r] = 0; }

    // --- scan all 1024 codes, 16 per WMMA tile ------------------------------
#pragma unroll 4
    for (int kt = 0; kt < 64; ++kt) {
      float2 bl = sBlo[(kt << 5) + lane];
      float2 bh = sBhi[(kt << 5) + lane];
      float  wn = sWn[(kt << 4) + m];
      v2f blo; blo.x = bl.x; blo.y = bl.y;
      v2f bhi; bhi.x = bh.x; bhi.y = bh.y;
      v8f zero = {0.f, 0.f, 0.f, 0.f, 0.f, 0.f, 0.f, 0.f};
      // dot = z . w   (D split 8 = 4 + 4, C-chained)
      v8f acc = __builtin_amdgcn_wmma_f32_16x16x4_f32(
          false, alo, false, blo, (short)0, zero, false, false);
      acc = __builtin_amdgcn_wmma_f32_16x16x4_f32(
          false, ahi, false, bhi, (short)0, acc, false, false);
      const int kcol = (kt << 4) + m;
#pragma unroll
      for (int r = 0; r < 8; ++r) {
        float cand = fmaf(acc[r], -2.0f, wn);  // ||w||^2 - 2 z.w
        bool lt = cand < minv[r];
        minv[r] = lt ? cand : minv[r];
        mini[r] = lt ? kcol : mini[r];
      }
    }

    // --- argmin across the 16 lanes of each column group --------------------
#pragma unroll
    for (int s = 8; s >= 1; s >>= 1) {
#pragma unroll
      for (int r = 0; r < 8; ++r) {
        float ov = __shfl_xor(minv[r], s, 32);
        int   oi = __shfl_xor(mini[r], s, 32);
        bool lt = ov < minv[r];
        minv[r] = lt ? ov : minv[r];
        mini[r] = lt ? oi : mini[r];
      }
    }

    // rows 0-7 live in lane 0, rows 8-15 in lane 16; park in per-wave LDS
    const int wsb = wib << 4;
    if (lane == 0) {
#pragma unroll
      for (int r = 0; r < 8; ++r) { sIdx[wsb + r] = mini[r]; sMin[wsb + r] = minv[r]; }
    }
    if (lane == 16) {
#pragma unroll
      for (int r = 0; r < 8; ++r) { sIdx[wsb + 8 + r] = mini[r]; sMin[wsb + 8 + r] = minv[r]; }
    }
    // intra-wave LDS RAW: DS ops are in-order per wave; compiler inserts dscnt wait

    float cont = 0.0f;
    if (lane < 16) {
      const int   myIdx = sIdx[wsb + lane];
      const float myMin = sMin[wsb + lane];
      cont = rowNorm + myMin;                 // ||z - w_idx||^2

      // gather the winning code from the shuffled LDS codebook
      int gb = ((myIdx >> 4) << 5) + (myIdx & 15);
      float2 c01 = sBlo[gb], c23 = sBlo[gb + 16];
      float2 c45 = sBhi[gb], c67 = sBhi[gb + 16];

      float* q = out + (((size_t)(b * 8) * 128 + h) * 128 + (w0 + lane));
      q[0]        = c01.x;
      q[HW]       = c01.y;
      q[2 * HW]   = c23.x;
      q[3 * HW]   = c23.y;
      q[4 * HW]   = c45.x;
      q[5 * HW]   = c45.y;
      q[6 * HW]   = c67.x;
      q[7 * HW]   = c67.y;
      out[IDX_OFF + n0 + lane] = (float)myIdx;
    }
    // wave-level sum of loss contributions (deterministic within wave)
#pragma unroll
    for (int s = 16; s >= 1; s >>= 1) cont += __shfl_xor(cont, s, 32);
    if (lane == 0) waveLoss += cont;
  }

  // --- deterministic block reduction (fixed order) ---------------------------
  if (lane == 0) sLoss[wib] = waveLoss;
  __syncthreads();
  if (tid == 0) {
    float s = 0.0f;
#pragma unroll
    for (int i = 0; i < WPB; ++i) s += sLoss[i];
    blockLoss[blockIdx.x] = s;
  }
}

__global__ __launch_bounds__(256) void vq_finalize_kernel(
    const float* __restrict__ blockLoss, float* __restrict__ out) {
  __shared__ float red[256];
  const int t = threadIdx.x;
  float s = blockLoss[t] + blockLoss[t + 256] + blockLoss[t + 512] + blockLoss[t + 768];
  red[t] = s;
  __syncthreads();
#pragma unroll
  for (int off = 128; off >= 1; off >>= 1) {
    if (t < off) red[t] += red[t + off];
    __syncthreads();
  }
  if (t == 0)
    out[LOSS_OFF] = 0.25f * red[0] / (float)((size_t)N_ROWS * EMB_DIM);
}

extern "C" void kernel_launch(void* const* d_in, const int* in_sizes, int n_in,
                              void* d_out, int out_size, void* d_ws, size_t ws_size,
                              hipStream_t stream) {
  const float* z   = (const float*)d_in[0];  // (16, 8, 128, 128) f32
  const float* wgt = (const float*)d_in[1];  // (1024, 8) f32
  float* out = (float*)d_out;
  float* blockLoss = (float*)d_ws;           // 1024 f32 partials (overwritten each call)

  vq_main_kernel<<<BLOCKS, TPB, 0, stream>>>(z, wgt, out, blockLoss);
  vq_finalize_kernel<<<1, 256, 0, stream>>>(blockLoss, out);
}